// EncoderBlock_87462714015831
// MI455X (gfx1250) — compile-verified
//
#include <hip/hip_runtime.h>
#include <hip/hip_bf16.h>

#define D_MODEL 1024
#define N_HEAD  8
#define D_HEAD  128
#define SEQ     2048
#define BATCH   4
#define D_FF    4096
#define ROWS    (BATCH * SEQ)   // 8192
#define GK      16              // GEMM K-step staged through LDS
#define PSTR    160             // dwords per k-pair row: 160 % 64 = 32 -> conflict-free halves

typedef __attribute__((ext_vector_type(2))) float v2f;
typedef __attribute__((ext_vector_type(8))) float v8f;

__device__ __forceinline__ v8f wmma4(v2f a, v2f b, v8f c) {
  // V_WMMA_F32_16X16X4_F32: D = A(16x4) * B(4x16) + C(16x16), all fp32
  return __builtin_amdgcn_wmma_f32_16x16x4_f32(
      /*neg_a=*/false, a, /*neg_b=*/false, b,
      /*c_mod=*/(short)0, c, /*reuse_a=*/false, /*reuse_b=*/false);
}

// ---------------------------------------------------------------------------
// GEMM: C[. x N] = A[. x K] @ W[K x N] + bias (+optional ReLU)
// block = 256 threads = 8 waves; block tile 128(M) x 64(N).
// W tile (16x64) staged in double-buffered LDS, k-pair interleaved:
//   wt[pair][col][j] = W[k0 + 2*pair + j][bcol + col]
// so each WMMA B fragment is ONE aligned ds_load_b64 (no register repacking).
// grid = (N/64, M/128).
// ---------------------------------------------------------------------------
__global__ __launch_bounds__(256) void gemm_bias_kernel(
    const float* __restrict__ A, const float* __restrict__ W,
    const float* __restrict__ bias, float* __restrict__ C,
    int K, int N, int relu)
{
  __shared__ float wt[2][8 * PSTR];   // 8 k-pairs per 16-k step, padded rows
  const int tid  = threadIdx.x;
  const int lane = tid & 31;
  const int wave = tid >> 5;
  const int n    = lane & 15;       // also A-row sub-index m
  const int hi   = lane >> 4;
  const int koff = hi * 2;
  const int mtile = blockIdx.y * 8 + wave;
  const int bcol  = blockIdx.x * 64;
  const int row   = mtile * 16 + n;
  const float* __restrict__ arow = A + (size_t)row * K;

  // Cooperative W-tile loader: thread -> (k-pair tid/32, col pair (tid&31)*2)
  const int p  = tid >> 5;            // 0..7
  const int cp = (lane) * 2;          // 0,2,...,62
  const float* __restrict__ wp = W + (size_t)(2 * p) * N + bcol + cp;
  const size_t wstep = (size_t)GK * N;
  const int ldst = p * PSTR + cp * 2; // dword offset of this thread's float4

  v2f r0 = *(const v2f*)wp;           // row 2p,   cols cp..cp+1
  v2f r1 = *(const v2f*)(wp + N);     // row 2p+1, cols cp..cp+1
  wp += wstep;
  float4 packed = make_float4(r0.x, r1.x, r0.y, r1.y);
  *(float4*)&wt[0][ldst] = packed;
  __syncthreads();

  v8f acc[4] = {{}, {}, {}, {}};
  const int nsteps = K / GK;
  for (int step = 0; step < nsteps; ++step) {
    const int buf = step & 1;
    if (step + 1 < nsteps) {          // prefetch next W tile into registers
      r0 = *(const v2f*)wp;
      r1 = *(const v2f*)(wp + N);
      wp += wstep;
    }
    // A fragments for this K-step (contiguous 8B loads)
    v2f av[4];
#pragma unroll
    for (int c = 0; c < 4; ++c)
      av[c] = *(const v2f*)(arow + step * GK + 4 * c + koff);

#pragma unroll
    for (int c = 0; c < 4; ++c) {
      const float* bsrc = &wt[buf][(2 * c + hi) * PSTR + n * 2];
#pragma unroll
      for (int t = 0; t < 4; ++t) {
        v2f bf = *(const v2f*)(bsrc + t * 32);  // one ds_load_b64
        acc[t] = wmma4(av[c], bf, acc[t]);
      }
    }
    __syncthreads();                  // all reads of wt[buf] done
    if (step + 1 < nsteps)
      *(float4*)&wt[buf ^ 1][ldst] = make_float4(r0.x, r1.x, r0.y, r1.y);
    __syncthreads();                  // next tile visible
  }

#pragma unroll
  for (int t = 0; t < 4; ++t) {
    const int col = bcol + t * 16 + n;
    const float bv = bias[col];
#pragma unroll
    for (int v = 0; v < 8; ++v) {
      float val = acc[t][v] + bv;
      if (relu) val = fmaxf(val, 0.0f);
      C[(size_t)(mtile * 16 + v + 8 * hi) * N + col] = val;
    }
  }
}

// ---------------------------------------------------------------------------
// Flash attention, one wave per 16-query tile per (b,h).
// Q,K,V stored [B*S, D] with head slice at h*D_HEAD.
// grid = (S/16/8, H, B), block = 256 (8 waves). LDS: 1KB/wave for P reshape.
// ---------------------------------------------------------------------------
__global__ __launch_bounds__(256) void attn_kernel(
    const float* __restrict__ Q, const float* __restrict__ Km,
    const float* __restrict__ V, float* __restrict__ ctx)
{
  __shared__ float lds[8 * 256];
  const int lane  = threadIdx.x & 31;
  const int wave  = threadIdx.x >> 5;
  const int h     = blockIdx.y;
  const int b     = blockIdx.z;
  const int qt    = blockIdx.x * 8 + wave;
  const int qbase = qt * 16;
  const int n     = lane & 15;
  const int hi    = lane >> 4;
  const int koff  = hi * 2;
  const size_t hoff = (size_t)h * D_HEAD;
  float* __restrict__ pl = lds + wave * 256;

  // Preload Q tile as 32 K=4 A-fragments (row m = n, k = 4c + koff + {0,1})
  const float* __restrict__ qrow =
      Q + ((size_t)(b * SEQ + qbase + n)) * D_MODEL + hoff;
  v2f qreg[32];
#pragma unroll
  for (int c = 0; c < 32; ++c)
    qreg[c] = *(const v2f*)(qrow + 4 * c + koff);

  v8f out[8];
#pragma unroll
  for (int t = 0; t < 8; ++t) out[t] = {};
  float mrow[8], lrow[8];
#pragma unroll
  for (int v = 0; v < 8; ++v) { mrow[v] = -3.0e38f; lrow[v] = 0.0f; }

  const float scale = 0.08838834764831845f;  // 1/sqrt(128)
  const float* __restrict__ krow = Km + ((size_t)(b * SEQ + n)) * D_MODEL + hoff;
  const float* __restrict__ vrow = V + ((size_t)(b * SEQ)) * D_MODEL + hoff;
  const size_t tstep = (size_t)16 * D_MODEL;  // advance 16 keys

  for (int kt = 0; kt < SEQ / 16; ++kt) {
    // ---- S = Q @ K^T  (B-fragment: n = key index, k-dim = d, contiguous) ----
    v8f s = {};
#pragma unroll
    for (int g = 0; g < 4; ++g) {
      v2f kf[8];
#pragma unroll
      for (int c = 0; c < 8; ++c)
        kf[c] = *(const v2f*)(krow + 4 * (8 * g + c) + koff);
#pragma unroll
      for (int c = 0; c < 8; ++c)
        s = wmma4(qreg[8 * g + c], kf[c], s);
    }

    // ---- online softmax (C-layout rows live in 16-lane groups) ----
    float alpha[8];
#pragma unroll
    for (int v = 0; v < 8; ++v) {
      float sv = s[v] * scale;
      float mx = sv;
      mx = fmaxf(mx, __shfl_xor(mx, 1, 32));
      mx = fmaxf(mx, __shfl_xor(mx, 2, 32));
      mx = fmaxf(mx, __shfl_xor(mx, 4, 32));
      mx = fmaxf(mx, __shfl_xor(mx, 8, 32));
      const float mnew = fmaxf(mrow[v], mx);
      const float p = __expf(sv - mnew);
      float ps = p;
      ps += __shfl_xor(ps, 1, 32);
      ps += __shfl_xor(ps, 2, 32);
      ps += __shfl_xor(ps, 4, 32);
      ps += __shfl_xor(ps, 8, 32);
      const float al = __expf(mrow[v] - mnew);
      lrow[v] = lrow[v] * al + ps;
      mrow[v] = mnew;
      alpha[v] = al;
      pl[(v + 8 * hi) * 16 + n] = p;  // P row-major [16][16] in LDS
    }
    asm volatile("s_wait_dscnt 0" ::: "memory");  // wave-local LDS RAW

#pragma unroll
    for (int t = 0; t < 8; ++t)
#pragma unroll
      for (int v = 0; v < 8; ++v)
        out[t][v] *= alpha[v];

    // Reload P as A-fragments (C-layout -> A-layout via LDS)
    v2f pa[4];
#pragma unroll
    for (int c2 = 0; c2 < 4; ++c2) {
      pa[c2].x = pl[n * 16 + 4 * c2 + koff];
      pa[c2].y = pl[n * 16 + 4 * c2 + koff + 1];
    }

    // ---- out += P(16x16) @ V(16x128) ----
#pragma unroll
    for (int c2 = 0; c2 < 4; ++c2) {
      const float* v0 = vrow + (size_t)(4 * c2 + koff) * D_MODEL + n;
      const float* v1 = v0 + D_MODEL;
      v2f vf[8];
#pragma unroll
      for (int t = 0; t < 8; ++t) { vf[t].x = v0[t * 16]; vf[t].y = v1[t * 16]; }
#pragma unroll
      for (int t = 0; t < 8; ++t)
        out[t] = wmma4(pa[c2], vf[t], out[t]);
    }

    krow += tstep;
    vrow += tstep;
  }

#pragma unroll
  for (int t = 0; t < 8; ++t) {
#pragma unroll
    for (int v = 0; v < 8; ++v) {
      const size_t orow = (size_t)(b * SEQ) + qbase + v + 8 * hi;
      ctx[orow * D_MODEL + hoff + t * 16 + n] = out[t][v] / lrow[v];
    }
  }
}

// ---------------------------------------------------------------------------
// out = LayerNorm(a + b) * g + beta, one row (D=1024) per 256-thread block
// ---------------------------------------------------------------------------
__global__ __launch_bounds__(256) void add_ln_kernel(
    const float* __restrict__ a, const float* __restrict__ b,
    const float* __restrict__ g, const float* __restrict__ beta,
    float* __restrict__ out)
{
  __shared__ float sh[16];
  const int row  = blockIdx.x;
  const int tid  = threadIdx.x;
  const int lane = tid & 31;
  const int wave = tid >> 5;
  const float* pa = a + (size_t)row * D_MODEL;
  const float* pb = b + (size_t)row * D_MODEL;

  float x[4];
  float s = 0.0f, ss = 0.0f;
#pragma unroll
  for (int i = 0; i < 4; ++i) {
    const float v = pa[tid + 256 * i] + pb[tid + 256 * i];
    x[i] = v;
    s += v;
    ss += v * v;
  }
#pragma unroll
  for (int m = 1; m < 32; m <<= 1) {
    s  += __shfl_xor(s, m, 32);
    ss += __shfl_xor(ss, m, 32);
  }
  if (lane == 0) { sh[wave] = s; sh[wave + 8] = ss; }
  __syncthreads();
  float ts = 0.0f, tss = 0.0f;
#pragma unroll
  for (int w = 0; w < 8; ++w) { ts += sh[w]; tss += sh[w + 8]; }
  const float mu  = ts * (1.0f / D_MODEL);
  const float var = tss * (1.0f / D_MODEL) - mu * mu;
  const float inv = rsqrtf(var + 1e-6f);
#pragma unroll
  for (int i = 0; i < 4; ++i) {
    const int col = tid + 256 * i;
    out[(size_t)row * D_MODEL + col] = (x[i] - mu) * inv * g[col] + beta[col];
  }
}

// ---------------------------------------------------------------------------
extern "C" void kernel_launch(void* const* d_in, const int* in_sizes, int n_in,
                              void* d_out, int out_size, void* d_ws, size_t ws_size,
                              hipStream_t stream) {
  const float* x     = (const float*)d_in[0];
  const float* Wq    = (const float*)d_in[1];
  const float* bq    = (const float*)d_in[2];
  const float* Wk    = (const float*)d_in[3];
  const float* bk    = (const float*)d_in[4];
  const float* Wv    = (const float*)d_in[5];
  const float* bv    = (const float*)d_in[6];
  const float* ln1_g = (const float*)d_in[7];
  const float* ln1_b = (const float*)d_in[8];
  const float* W1    = (const float*)d_in[9];
  const float* b1    = (const float*)d_in[10];
  const float* W2    = (const float*)d_in[11];
  const float* b2    = (const float*)d_in[12];
  const float* ln2_g = (const float*)d_in[13];
  const float* ln2_b = (const float*)d_in[14];
  float* outp = (float*)d_out;

  float* ws = (float*)d_ws;
  const size_t NTOK = (size_t)ROWS * D_MODEL;  // 8M floats
  float* Qb = ws;             // [ROWS, D]
  float* Kb = ws + NTOK;      // [ROWS, D]
  float* Vb = ws + 2 * NTOK;  // [ROWS, D]
  float* Cx = ws + 3 * NTOK;  // [ROWS, D] attention context
  float* X1 = ws + 4 * NTOK;  // [ROWS, D] after LN1 (persists)
  float* H2 = ws + 5 * NTOK;  // [ROWS, D] FFN output
  float* H1 = ws;             // [ROWS, DFF] reuses Q/K/V/Cx region (consumed)

  dim3 blk(256);
  dim3 gD (D_MODEL / 64, ROWS / 128);  // N=1024 GEMMs
  dim3 gF (D_FF    / 64, ROWS / 128);  // N=4096 GEMM

  // QKV projections
  gemm_bias_kernel<<<gD, blk, 0, stream>>>(x, Wq, bq, Qb, D_MODEL, D_MODEL, 0);
  gemm_bias_kernel<<<gD, blk, 0, stream>>>(x, Wk, bk, Kb, D_MODEL, D_MODEL, 0);
  gemm_bias_kernel<<<gD, blk, 0, stream>>>(x, Wv, bv, Vb, D_MODEL, D_MODEL, 0);

  // Multi-head flash attention
  attn_kernel<<<dim3(SEQ / 128, N_HEAD, BATCH), blk, 0, stream>>>(Qb, Kb, Vb, Cx);

  // x1 = LN(ctx + x)
  add_ln_kernel<<<ROWS, blk, 0, stream>>>(Cx, x, ln1_g, ln1_b, X1);

  // FFN
  gemm_bias_kernel<<<gF, blk, 0, stream>>>(X1, W1, b1, H1, D_MODEL, D_FF, 1);
  gemm_bias_kernel<<<gD, blk, 0, stream>>>(H1, W2, b2, H2, D_FF, D_MODEL, 0);

  // out = LN(h + x1)
  add_ln_kernel<<<ROWS, blk, 0, stream>>>(H2, X1, ln2_g, ln2_b, outp);
}